// ProbSparseAttention_85942295593272
// MI455X (gfx1250) — compile-verified
//
#include <hip/hip_runtime.h>
#include <math.h>

typedef float v2f __attribute__((ext_vector_type(2)));
typedef float v8f __attribute__((ext_vector_type(8)));

#define B_    4
#define L_    2048
#define H_    8
#define D_    64
#define U_    40
#define UPAD_ 48
#define BH_   (B_ * H_)
#define KSPLIT_ 8
#define SCALE_ 0.125f   /* 1/sqrt(64) */

// Deterministic stand-in for jax.random.randint(key(42), (L,40), 0, L).
// Shared across (b,h) like the reference (indices depend only on (q,s)).
__device__ __forceinline__ unsigned sample_idx(unsigned q, unsigned s) {
    unsigned x = q * 40u + s + 0x9E3779B9u;
    x ^= x >> 16; x *= 0x85EBCA6Bu;
    x ^= x >> 13; x *= 0xC2B2AE35u;
    x ^= x >> 16;
    return x & (L_ - 1u);
}

// ---------------- Pass 1: sparsity measure M = max - mean over 40 sampled keys
__global__ __launch_bounds__(256) void k_measure(const float* __restrict__ Q,
                                                 const float* __restrict__ K,
                                                 float* __restrict__ M) {
    int tid  = blockIdx.x * 256 + threadIdx.x;
    int w    = tid >> 5;          // one wave per query
    int lane = tid & 31;
    if (w >= BH_ * L_) return;
    int bh = w / L_;
    int q  = w - bh * L_;
    int b = bh >> 3, h = bh & 7;
    const float* qrow = Q + (((size_t)(b * L_ + q)) * H_ + h) * D_;
    float q0 = qrow[lane], q1 = qrow[32 + lane];
    float mx = -1e30f, sm = 0.f;
    for (int s = 0; s < U_; ++s) {
        int idx = (int)sample_idx((unsigned)q, (unsigned)s);
        const float* krow = K + (((size_t)(b * L_ + idx)) * H_ + h) * D_;
        float p = q0 * krow[lane] + q1 * krow[32 + lane];
        #pragma unroll
        for (int off = 16; off > 0; off >>= 1) p += __shfl_xor(p, off, 32);
        mx = fmaxf(mx, p);
        sm += p;
    }
    if (lane == 0) M[(size_t)bh * L_ + q] = mx - sm * (1.0f / L_);
}

// ---------------- Pass 2: top-40 queries per (b,h)
__global__ __launch_bounds__(256) void k_topk(const float* __restrict__ M,
                                              int* __restrict__ Mtop) {
    __shared__ float vals[L_];
    __shared__ float bv[256];
    __shared__ int   bi[256];
    int bh = blockIdx.x;
    int t  = threadIdx.x;
    for (int i = t; i < L_; i += 256) vals[i] = M[(size_t)bh * L_ + i];
    __syncthreads();
    for (int it = 0; it < U_; ++it) {
        float best = -1e30f; int bidx = 0;
        for (int i = t; i < L_; i += 256) {
            float v = vals[i];
            if (v > best) { best = v; bidx = i; }
        }
        bv[t] = best; bi[t] = bidx;
        __syncthreads();
        for (int s = 128; s > 0; s >>= 1) {
            if (t < s && bv[t + s] > bv[t]) { bv[t] = bv[t + s]; bi[t] = bi[t + s]; }
            __syncthreads();
        }
        if (t == 0) { Mtop[bh * UPAD_ + it] = bi[0]; vals[bi[0]] = -1e30f; }
        __syncthreads();
    }
    if (t >= U_ && t < UPAD_) Mtop[bh * UPAD_ + t] = -1;   // padding rows
}

// ---------------- Gather selected Q rows (zero-padded to 48)
__global__ void k_gather_q(const float* __restrict__ Q,
                           const int* __restrict__ Mtop,
                           float* __restrict__ Qsel) {
    int bu = blockIdx.x;
    int bh = bu / UPAD_, u = bu - bh * UPAD_;
    int b = bh >> 3, h = bh & 7;
    int q = Mtop[bh * UPAD_ + u];
    int d = threadIdx.x;
    float v = 0.f;
    if (q >= 0) v = Q[(((size_t)(b * L_ + q)) * H_ + h) * D_ + d];
    Qsel[(size_t)bu * D_ + d] = v;
}

// ---------------- V column means per (b,h)
__global__ void k_vmean(const float* __restrict__ V, float* __restrict__ Vmean) {
    int bh = blockIdx.x;
    int b = bh >> 3, h = bh & 7;
    int d = threadIdx.x;
    const float* base = V + ((size_t)b * L_ * H_ + h) * D_ + d;
    float s = 0.f;
    for (int l = 0; l < L_; ++l) s += base[(size_t)l * (H_ * D_)];
    Vmean[bh * D_ + d] = s * (1.0f / L_);
}

// ---------------- Broadcast-fill output with mean(V)
__global__ void k_fill(const float* __restrict__ Vmean, float* __restrict__ out, int n) {
    int i = blockIdx.x * 256 + threadIdx.x;
    if (i >= n) return;
    int hd = i & (H_ * D_ - 1);           // h*D + d
    int b  = i >> 20;                     // / (L*H*D) = / 2^20
    out[i] = Vmean[b * (H_ * D_) + hd];
}

// ---------------- scores = Qsel(48x64) @ K^T (64x2048), WMMA f32 16x16x4
__global__ __launch_bounds__(256) void k_scores(const float* __restrict__ Qsel,
                                                const float* __restrict__ K,
                                                float* __restrict__ scores) {
    int g = blockIdx.x;
    int split = g & 3; g >>= 2;
    int mt = g % 3;
    int bh = g / 3;
    int b = bh >> 3, h = bh & 7;
    int wave = threadIdx.x >> 5, lane = threadIdx.x & 31;
    int m_base = mt * 16;
    int doff = (lane < 16) ? 0 : 2;
    int mrow = m_base + (lane & 15);
    // Cache A (16x64 slab) in registers: 16 k-steps of (K0,K1)/(K2,K3) pairs.
    v2f a[16];
    const float* qrow = Qsel + (size_t)(bh * UPAD_ + mrow) * D_;
    #pragma unroll
    for (int kk = 0; kk < 16; ++kk)
        a[kk] = *reinterpret_cast<const v2f*>(qrow + kk * 4 + doff);
    for (int nt = split * 32 + wave; nt < split * 32 + 32; nt += 8) {
        int col = nt * 16 + (lane & 15);
        const float* krow = K + (((size_t)(b * L_ + col)) * H_ + h) * D_;
        v8f c = {};
        #pragma unroll
        for (int kk = 0; kk < 16; ++kk) {
            v2f bb = *reinterpret_cast<const v2f*>(krow + kk * 4 + doff);
            c = __builtin_amdgcn_wmma_f32_16x16x4_f32(false, a[kk], false, bb,
                                                      (short)0, c, false, false);
        }
        #pragma unroll
        for (int r = 0; r < 8; ++r) {
            int rr = m_base + r + ((lane < 16) ? 0 : 8);
            scores[(size_t)(bh * UPAD_ + rr) * L_ + nt * 16 + (lane & 15)] = c[r] * SCALE_;
        }
    }
}

// ---------------- Row softmax over L_K=2048 (one block per row)
__global__ __launch_bounds__(256) void k_softmax(float* __restrict__ scores) {
    __shared__ float red[256];
    size_t base = (size_t)blockIdx.x * L_;
    int t = threadIdx.x;
    float x[8];
    float mx = -1e30f;
    #pragma unroll
    for (int i = 0; i < 8; ++i) { x[i] = scores[base + t + i * 256]; mx = fmaxf(mx, x[i]); }
    red[t] = mx; __syncthreads();
    for (int s = 128; s > 0; s >>= 1) { if (t < s) red[t] = fmaxf(red[t], red[t + s]); __syncthreads(); }
    mx = red[0]; __syncthreads();
    float sm = 0.f;
    #pragma unroll
    for (int i = 0; i < 8; ++i) { x[i] = __expf(x[i] - mx); sm += x[i]; }
    red[t] = sm; __syncthreads();
    for (int s = 128; s > 0; s >>= 1) { if (t < s) red[t] += red[t + s]; __syncthreads(); }
    float inv = 1.0f / red[0];
    #pragma unroll
    for (int i = 0; i < 8; ++i) scores[base + t + i * 256] = x[i] * inv;
}

// ---------------- ctx_part = attn(48x2048) @ V(2048x64), K-split x8, WMMA f32
__global__ __launch_bounds__(128) void k_ctx(const float* __restrict__ attn,
                                             const float* __restrict__ V,
                                             float* __restrict__ ctxpart) {
    int g = blockIdx.x;
    int split = g & (KSPLIT_ - 1); g >>= 3;
    int mt = g % 3;
    int bh = g / 3;
    int b = bh >> 3, h = bh & 7;
    int wave = threadIdx.x >> 5, lane = threadIdx.x & 31;
    int m_base = mt * 16;
    int doff = (lane < 16) ? 0 : 2;
    int mrow = m_base + (lane & 15);
    int col  = wave * 16 + (lane & 15);
    const float* arow = attn + (size_t)(bh * UPAD_ + mrow) * L_ + doff;
    const float* vcol = V + ((size_t)b * L_ * H_ + h) * D_ + col;
    int k_lo = split * (L_ / KSPLIT_), k_hi = k_lo + (L_ / KSPLIT_);
    v8f c = {};
    for (int k0 = k_lo; k0 < k_hi; k0 += 4) {
        v2f aa = *reinterpret_cast<const v2f*>(arow + k0);
        v2f bb;
        bb[0] = vcol[(size_t)(k0 + doff) * (H_ * D_)];
        bb[1] = vcol[(size_t)(k0 + doff + 1) * (H_ * D_)];
        c = __builtin_amdgcn_wmma_f32_16x16x4_f32(false, aa, false, bb,
                                                  (short)0, c, false, false);
    }
    #pragma unroll
    for (int r = 0; r < 8; ++r) {
        int rr = m_base + r + ((lane < 16) ? 0 : 8);
        ctxpart[((size_t)split * (BH_ * UPAD_) + bh * UPAD_ + rr) * D_ + col] = c[r];
    }
}

// ---------------- Sum K-split partials and scatter into selected output rows
__global__ void k_scatter(const float* __restrict__ ctxpart,
                          const int* __restrict__ Mtop,
                          float* __restrict__ out) {
    int g = blockIdx.x;
    int bh = g / U_, u = g - bh * U_;
    int b = bh >> 3, h = bh & 7;
    int q = Mtop[bh * UPAD_ + u];
    if (q < 0) return;
    int d = threadIdx.x;
    float s = 0.f;
    #pragma unroll
    for (int p = 0; p < KSPLIT_; ++p)
        s += ctxpart[((size_t)p * (BH_ * UPAD_) + bh * UPAD_ + u) * D_ + d];
    out[(((size_t)(b * L_ + q)) * H_ + h) * D_ + d] = s;
}

extern "C" void kernel_launch(void* const* d_in, const int* in_sizes, int n_in,
                              void* d_out, int out_size, void* d_ws, size_t ws_size,
                              hipStream_t stream) {
    const float* Q = (const float*)d_in[0];
    const float* K = (const float*)d_in[1];
    const float* V = (const float*)d_in[2];
    float* out = (float*)d_out;
    float* ws  = (float*)d_ws;

    // workspace layout (float elements)
    float* M       = ws;                                   // 32*2048      = 65536
    int*   Mtop    = (int*)(ws + 65536);                   // 32*48        = 1536
    float* Vmean   = ws + 67072;                           // 32*64        = 2048
    float* Qsel    = ws + 69120;                           // 32*48*64     = 98304
    float* scores  = ws + 167424;                          // 32*48*2048   = 3145728
    float* ctxpart = ws + 3313152;                         // 8*32*48*64   = 786432
    // total 4,099,584 floats ~= 16.4 MB

    const int nOut = B_ * L_ * H_ * D_;                    // 4,194,304

    k_measure <<< (BH_ * L_) / 8,        256, 0, stream >>> (Q, K, M);
    k_topk    <<< BH_,                   256, 0, stream >>> (M, Mtop);
    k_gather_q<<< BH_ * UPAD_,            64, 0, stream >>> (Q, Mtop, Qsel);
    k_vmean   <<< BH_,                    64, 0, stream >>> (V, Vmean);
    k_fill    <<< (nOut + 255) / 256,    256, 0, stream >>> (Vmean, out, nOut);
    k_scores  <<< BH_ * 3 * 4,           256, 0, stream >>> (Qsel, K, scores);
    k_softmax <<< BH_ * UPAD_,           256, 0, stream >>> (scores);
    k_ctx     <<< BH_ * 3 * KSPLIT_,     128, 0, stream >>> (scores, V, ctxpart);
    k_scatter <<< BH_ * U_,               64, 0, stream >>> (ctxpart, Mtop, out);
}